// EnvLight_88622355186065
// MI455X (gfx1250) — compile-verified
//
#include <hip/hip_runtime.h>

#define HPIX 1080
#define WPIX 1920
#define NPIX (HPIX * WPIX)
#define NMIP 8
#define NCH  3
#define MH   128
#define MW   256
#define PLANE (MH * MW)

// gfx1250 async-to-LDS path (ASYNCcnt-tracked DMA), guarded so compilation
// falls back to plain loads if this toolchain lacks the builtins.
#if defined(__gfx1250__) && \
    __has_builtin(__builtin_amdgcn_global_load_async_to_lds_b32) && \
    __has_builtin(__builtin_amdgcn_s_wait_asynccnt)
#define USE_ASYNC_LDS 1
#else
#define USE_ASYNC_LDS 0
#endif

// ---------------------------------------------------------------------------
// Kernel 1: repack mips (M,C,H,W) planar -> (M,H,W,4) interleaved float4 in ws.
// On gfx1250: stage the 3 channel planes into LDS via global_load_async_to_lds,
// s_wait_asynccnt, then emit one coalesced global_store_b128 per texel.
// ---------------------------------------------------------------------------
__global__ void __launch_bounds__(256) reorder_mips_kernel(
    const float* __restrict__ mips, float4* __restrict__ wsm) {
  int t = blockIdx.x * 256 + threadIdx.x;     // grid exactly covers NMIP*PLANE
  int m  = t / PLANE;
  int yx = t - m * PLANE;
  const float* p = mips + (size_t)m * NCH * PLANE + yx;
  float4 v;
#if USE_ASYNC_LDS
  __shared__ float lds[3 * 256];
  const unsigned tid = threadIdx.x;
  // Builtin signature (from hipcc diagnostic): arg0 is addrspace(1) int*,
  // arg1 is addrspace(3) int*, then imm offset, imm cpol.
  typedef __attribute__((address_space(1))) int* gint_p;
  typedef __attribute__((address_space(3))) int* lint_p;
  float* g0 = const_cast<float*>(p);
  float* g1 = const_cast<float*>(p + PLANE);
  float* g2 = const_cast<float*>(p + 2 * PLANE);
  __builtin_amdgcn_global_load_async_to_lds_b32(
      (gint_p)(void*)g0, (lint_p)(void*)&lds[tid], 0, 0);
  __builtin_amdgcn_global_load_async_to_lds_b32(
      (gint_p)(void*)g1, (lint_p)(void*)&lds[256 + tid], 0, 0);
  __builtin_amdgcn_global_load_async_to_lds_b32(
      (gint_p)(void*)g2, (lint_p)(void*)&lds[512 + tid], 0, 0);
  __builtin_amdgcn_s_wait_asynccnt(0);        // ASYNCcnt == 0 -> LDS visible
  v.x = lds[tid];
  v.y = lds[256 + tid];
  v.z = lds[512 + tid];
  v.w = 0.0f;
#else
  v.x = p[0];
  v.y = p[PLANE];
  v.z = p[2 * PLANE];
  v.w = 0.0f;
#endif
  wsm[t] = v;
}

__device__ __forceinline__ float3 tap_ws(const float4* __restrict__ wsm,
                                         int base, int x) {
  float4 t = wsm[(size_t)(base + x)];          // one global_load_b128
  return make_float3(t.x, t.y, t.z);
}

__device__ __forceinline__ float3 tap_pl(const float* __restrict__ mips,
                                         int base, int x) {
  const float* p = mips + (size_t)(base + x);
  return make_float3(p[0], p[PLANE], p[2 * PLANE]);
}

__device__ __forceinline__ float3 bilerp4(float3 t00, float3 t10,
                                          float3 t01, float3 t11,
                                          float w00, float w10,
                                          float w01, float w11) {
  float3 r;
  r.x = t00.x * w00 + t10.x * w10 + t01.x * w01 + t11.x * w11;
  r.y = t00.y * w00 + t10.y * w10 + t01.y * w01 + t11.y * w11;
  r.z = t00.z * w00 + t10.z * w10 + t01.z * w01 + t11.z * w11;
  return r;
}

// ---------------------------------------------------------------------------
// Kernel 2: per-pixel dir->uv, 2-mip bilinear gather (L2-resident), mip lerp.
// Grid-stride; global_prefetch_b8 pulls the next iteration's streaming inputs.
// ---------------------------------------------------------------------------
template <bool USE_WS>
__global__ void __launch_bounds__(256) envlight_kernel(
    const float*  __restrict__ xin,
    const float*  __restrict__ rough,
    const float*  __restrict__ mips,
    const float4* __restrict__ wsm,
    float*        __restrict__ out) {
  const int stride = gridDim.x * blockDim.x;
  for (int i = blockIdx.x * blockDim.x + threadIdx.x; i < NPIX; i += stride) {
    if (i + stride < NPIX) {
      // gfx1250: lowers to global_prefetch_b8
      __builtin_prefetch(xin + (size_t)(i + stride) * 3, 0, 0);
      __builtin_prefetch(rough + (size_t)(i + stride), 0, 0);
    }

    float dx = xin[3 * (size_t)i + 0];
    float dy = xin[3 * (size_t)i + 1];
    float dz = xin[3 * (size_t)i + 2];

    // ix = u*MW - 0.5 = phi*(MW/2pi) + 127.5 ; iy = theta*(MH/pi) - 0.5
    // (MW/(2pi) == MH/pi == 40.743665...)
    const float SC = 40.74366543152521f;
    float theta = acosf(fminf(fmaxf(dz, -1.0f), 1.0f));
    float phi   = atan2f(dy, dx);
    float ix = phi * SC + 127.5f;
    float iy = theta * SC - 0.5f;

    float fx0 = floorf(ix), fy0 = floorf(iy);
    float wx = ix - fx0, wy = iy - fy0;
    int xi0 = (int)fx0, yi0 = (int)fy0;
    int xi1 = min(max(xi0 + 1, 0), MW - 1);    // border padding = clamp
    int yi1 = min(max(yi0 + 1, 0), MH - 1);
    xi0 = min(max(xi0, 0), MW - 1);
    yi0 = min(max(yi0, 0), MH - 1);

    // roughness -> fractional mip index (mirror reference arithmetic)
    float r    = fminf(fmaxf(rough[i], 0.08f), 0.5f);
    float midx = (r - 0.08f) / 0.42f * (float)(NMIP - 1);
    float fidx = floorf(midx);
    int i0 = min(max((int)fidx, 0), NMIP - 2);
    int i1 = min(max((int)ceilf(midx), 0), NMIP - 1);
    float wm = midx - fidx;

    float w00 = (1.0f - wx) * (1.0f - wy);
    float w10 = wx * (1.0f - wy);
    float w01 = (1.0f - wx) * wy;
    float w11 = wx * wy;

    float3 c0, c1;
    if (USE_WS) {
      int b0r0 = (i0 * MH + yi0) * MW;   // row bases, shared by x0/x1 taps
      int b0r1 = (i0 * MH + yi1) * MW;
      int b1r0 = (i1 * MH + yi0) * MW;
      int b1r1 = (i1 * MH + yi1) * MW;
      c0 = bilerp4(tap_ws(wsm, b0r0, xi0), tap_ws(wsm, b0r0, xi1),
                   tap_ws(wsm, b0r1, xi0), tap_ws(wsm, b0r1, xi1),
                   w00, w10, w01, w11);
      c1 = bilerp4(tap_ws(wsm, b1r0, xi0), tap_ws(wsm, b1r0, xi1),
                   tap_ws(wsm, b1r1, xi0), tap_ws(wsm, b1r1, xi1),
                   w00, w10, w01, w11);
    } else {
      int b0r0 = (i0 * NCH * MH + yi0) * MW;
      int b0r1 = (i0 * NCH * MH + yi1) * MW;
      int b1r0 = (i1 * NCH * MH + yi0) * MW;
      int b1r1 = (i1 * NCH * MH + yi1) * MW;
      c0 = bilerp4(tap_pl(mips, b0r0, xi0), tap_pl(mips, b0r0, xi1),
                   tap_pl(mips, b0r1, xi0), tap_pl(mips, b0r1, xi1),
                   w00, w10, w01, w11);
      c1 = bilerp4(tap_pl(mips, b1r0, xi0), tap_pl(mips, b1r0, xi1),
                   tap_pl(mips, b1r1, xi0), tap_pl(mips, b1r1, xi1),
                   w00, w10, w01, w11);
    }

    out[3 * (size_t)i + 0] = c0.x + wm * (c1.x - c0.x);
    out[3 * (size_t)i + 1] = c0.y + wm * (c1.y - c0.y);
    out[3 * (size_t)i + 2] = c0.z + wm * (c1.z - c0.z);
  }
}

extern "C" void kernel_launch(void* const* d_in, const int* in_sizes, int n_in,
                              void* d_out, int out_size, void* d_ws, size_t ws_size,
                              hipStream_t stream) {
  const float* xin   = (const float*)d_in[0];   // (1080,1920,3) f32
  const float* rough = (const float*)d_in[1];   // (1080,1920,1) f32
  const float* mips  = (const float*)d_in[2];   // (8,3,128,256) f32
  float* out = (float*)d_out;

  const size_t ws_needed = (size_t)NMIP * PLANE * sizeof(float4);  // 4 MB
  const int main_blocks = 2048;  // grid-stride: ~4 pixels/thread, 8 wave32/block

  if (ws_size >= ws_needed) {
    float4* wsm = (float4*)d_ws;
    reorder_mips_kernel<<<(NMIP * PLANE) / 256, 256, 0, stream>>>(mips, wsm);
    envlight_kernel<true><<<main_blocks, 256, 0, stream>>>(xin, rough, mips, wsm, out);
  } else {
    envlight_kernel<false><<<main_blocks, 256, 0, stream>>>(xin, rough, mips, nullptr, out);
  }
}